// KDEGaussian_22952305230422
// MI455X (gfx1250) — compile-verified
//
#include <hip/hip_runtime.h>

// KDE Gaussian on MI455X (gfx1250, wave32).
// pairwise ||x-y||^2 -> exp -> row-sum, via V_WMMA_F32_16X16X4_F32 for the
// cross-term GEMM (L x B*N x 6, K padded to 8 = two K=4 WMMAs).

typedef __attribute__((ext_vector_type(2))) float v2f;
typedef __attribute__((ext_vector_type(8))) float v8f;

#define KDE_B 2
#define KDE_N 4096
#define KDE_D 6
#define KDE_L 2048
#define KDE_COLS (KDE_B * KDE_N)      // 8192
#define COLS_PER_WG 256               // 16 tiles of 16 cols; divides 4096 -> one b per chunk
#define N_CHUNKS (KDE_COLS / COLS_PER_WG)   // 32
#define L_TILES (KDE_L / 16)          // 128
// exp(-d2 / (2*0.5^2)) = exp(-2*d2) = exp2(-2*log2(e)*d2)
#define NEG2_LOG2E (-2.0f * 1.44269504088896340736f)

__global__ void kde_zero_ws(float* __restrict__ p, int n) {
    int i = blockIdx.x * blockDim.x + threadIdx.x;
    if (i < n) p[i] = 0.0f;
}

__global__ __launch_bounds__(32)
void kde_partial_kernel(const float* __restrict__ samples,    // [B,N,D] = [col, D]
                        const float* __restrict__ locations,  // [L, D]
                        float* __restrict__ partial)          // [L, B] accumulators
{
    const int lane = threadIdx.x;        // wave32
    const int hi   = lane >> 4;          // 0: lanes 0-15, 1: lanes 16-31
    const int li   = lane & 15;
    const int l0   = blockIdx.x * 16;    // location tile base row
    const int c0   = blockIdx.y * COLS_PER_WG;  // sample column base
    const int b    = c0 >> 12;           // c0 / 4096; constant over the chunk

    // ---- A matrix (16x4 f32) per-lane registers: M=li, K = 2*hi + {0,1} ----
    const float* lp = locations + (size_t)(l0 + li) * KDE_D;
    v2f a1, a2;
    a1.x = lp[2 * hi + 0];
    a1.y = lp[2 * hi + 1];
    // second WMMA covers K=4..7; dims 6,7 are zero padding
    a2.x = hi ? 0.0f : lp[4];
    a2.y = hi ? 0.0f : lp[5];

    // ---- per-row ||loc||^2 for the rows this lane accumulates (M = 8*hi + r) ----
    float ln[8];
#pragma unroll
    for (int r = 0; r < 8; ++r) {
        const float* q = locations + (size_t)(l0 + 8 * hi + r) * KDE_D;
        float s = 0.0f;
#pragma unroll
        for (int d = 0; d < KDE_D; ++d) { float v = q[d]; s += v * v; }
        ln[r] = s;
    }

    float acc[8];
#pragma unroll
    for (int r = 0; r < 8; ++r) acc[r] = 0.0f;

    // ---- loop over 16-column tiles ----
    for (int t = 0; t < COLS_PER_WG / 16; ++t) {
        const int c = c0 + t * 16 + li;             // this lane's column (B load + C/D col)
        const float* sp = samples + (size_t)c * KDE_D;

        // B matrix (4x16 f32): v0 = row K=2*hi, v1 = row K=2*hi+1, col N=li
        v2f b1, b2;
        b1.x = sp[2 * hi + 0];
        b1.y = sp[2 * hi + 1];
        b2.x = hi ? 0.0f : sp[4];
        b2.y = hi ? 0.0f : sp[5];

        // ||sample||^2 for this lane's column
        float sn = 0.0f;
#pragma unroll
        for (int d = 0; d < KDE_D; ++d) { float v = sp[d]; sn += v * v; }

        // cross[r][li] = sum_d loc[M=8*hi+r][d] * samp[c][d]
        v8f czero = {};
        v8f cr = __builtin_amdgcn_wmma_f32_16x16x4_f32(
            false, a1, false, b1, (short)0, czero, false, false);
        cr = __builtin_amdgcn_wmma_f32_16x16x4_f32(
            false, a2, false, b2, (short)0, cr, false, false);

#pragma unroll
        for (int r = 0; r < 8; ++r) {
            float d2 = ln[r] + sn - 2.0f * cr[r];
            acc[r] += __builtin_amdgcn_exp2f(d2 * NEG2_LOG2E);
        }
    }

    // ---- reduce the 16 columns held across lanes (within each 16-lane half) ----
#pragma unroll
    for (int m = 1; m < 16; m <<= 1) {
#pragma unroll
        for (int r = 0; r < 8; ++r)
            acc[r] += __shfl_xor(acc[r], m, 32);
    }

    if (li == 0) {
#pragma unroll
        for (int r = 0; r < 8; ++r) {
            int row = l0 + 8 * hi + r;
            atomicAdd(&partial[row * KDE_B + b], acc[r]);
        }
    }
}

__global__ __launch_bounds__(256)
void kde_normalize_kernel(const float* __restrict__ partial,  // [L, B]
                          float* __restrict__ out)            // [1, L, B]
{
    __shared__ float sm0[256];
    __shared__ float sm1[256];
    const int t = threadIdx.x;

    float s0 = 0.0f, s1 = 0.0f;
    for (int l = t; l < KDE_L; l += 256) {
        s0 += partial[l * KDE_B + 0];
        s1 += partial[l * KDE_B + 1];
    }
    sm0[t] = s0; sm1[t] = s1;
    __syncthreads();
    for (int st = 128; st > 0; st >>= 1) {
        if (t < st) { sm0[t] += sm0[t + st]; sm1[t] += sm1[t + st]; }
        __syncthreads();
    }
    const float inv0 = 1.0f / sm0[0];
    const float inv1 = 1.0f / sm1[0];

    for (int l = t; l < KDE_L; l += 256) {
        out[l * KDE_B + 0] = partial[l * KDE_B + 0] * inv0;
        out[l * KDE_B + 1] = partial[l * KDE_B + 1] * inv1;
    }
}

extern "C" void kernel_launch(void* const* d_in, const int* in_sizes, int n_in,
                              void* d_out, int out_size, void* d_ws, size_t ws_size,
                              hipStream_t stream) {
    const float* samples   = (const float*)d_in[0];   // [2,4096,6]
    const float* locations = (const float*)d_in[1];   // [2048,6]
    float* out     = (float*)d_out;                   // [1,2048,2]
    float* partial = (float*)d_ws;                    // [2048,2] f32 accumulators

    const int pn = KDE_L * KDE_B;                     // 4096
    kde_zero_ws<<<(pn + 255) / 256, 256, 0, stream>>>(partial, pn);

    dim3 grid(L_TILES, N_CHUNKS);                     // 128 x 32 waves
    kde_partial_kernel<<<grid, 32, 0, stream>>>(samples, locations, partial);

    kde_normalize_kernel<<<1, 256, 0, stream>>>(partial, out);
}